// NeighborSearch_67044439490942
// MI455X (gfx1250) — compile-verified
//
#include <hip/hip_runtime.h>
#include <hip/hip_bf16.h>

// CDNA5 / gfx1250 radius neighbor search.
//  d2 = |q|^2 - 2 q.d + |d|^2  via V_WMMA_F32_16X16X4_F32 (K=3 padded to 4, fp32).
//  Counting + compaction via wave32 ballot masks, SALU popcounts, scalar cursors.
//  Data-tile loads are software-pipelined 2 tiles deep to hide L2 latency
//  (only ~1 wave/SIMD at this grid size, so the wave must self-overlap).

typedef __attribute__((ext_vector_type(2))) float v2f;
typedef __attribute__((ext_vector_type(8))) float v8f;

#define TILE 16
#define WAVES_PER_BLOCK 8
#define MAX_TOTAL (16384 * 32)

__device__ __forceinline__ float lane_bcast_f(float v, int srcLane) {
    int i = __builtin_amdgcn_ds_bpermute(srcLane << 2, __float_as_int(v));
    return __int_as_float(i);
}

// Build the A operand (queries, scaled by -2, K padded 3->4) and the C seed (|q|^2).
// A layout (16x4 f32): lanes 0-15 -> {K0,K1}, lanes 16-31 -> {K2,K3}; M = lane&15.
// C/D layout: VGPR r holds row (r + 8*(lane>=16)), col = lane&15.
__device__ __forceinline__ void load_query_tile(const float* __restrict__ q,
                                                int row_base, unsigned lane,
                                                v2f& a, v8f& cinit) {
    int m = row_base + (int)(lane & 15);
    float qx = q[m * 3 + 0];
    float qy = q[m * 3 + 1];
    float qz = q[m * 3 + 2];
    float qq = qx * qx + qy * qy + qz * qz;   // replicated in both lane halves
    if (lane < 16) { a.x = -2.0f * qx; a.y = -2.0f * qy; }
    else           { a.x = -2.0f * qz; a.y = 0.0f; }
    int hi = (lane >= 16) ? 8 : 0;
#pragma unroll
    for (int r = 0; r < 8; ++r) cinit[r] = lane_bcast_f(qq, r + hi);
}

// One 16-point data tile staged in registers (per-lane column coordinates).
struct DTile { float x, y, z; };

__device__ __forceinline__ DTile dload(const float* __restrict__ data,
                                       int col_base, unsigned lane) {
    int n = col_base + (int)(lane & 15);
    DTile t;
    t.x = data[n * 3 + 0];
    t.y = data[n * 3 + 1];
    t.z = data[n * 3 + 2];
    return t;
}

// Compute the 8 ballot masks for one 16x16 tile of the distance matrix.
// mask[r]: bits 0-15 -> row (row_base+r), bits 16-31 -> row (row_base+r+8).
__device__ __forceinline__ void tile_masks(const DTile& t, unsigned lane,
                                           const v2f& a, const v8f& cinit,
                                           float r2, unsigned mask[8]) {
    v2f b;
    if (lane < 16) { b.x = t.x; b.y = t.y; }   // K0,K1
    else           { b.x = t.z; b.y = 0.0f; }  // K2,K3(pad)
    float dd = t.x * t.x + t.y * t.y + t.z * t.z; // lane-uniform per column
    float thresh = r2 - dd;                       // D[r] <= r2 - |d|^2 <=> d2 <= r2

    // D = (-2 q) . d + |q|^2
    v8f d = __builtin_amdgcn_wmma_f32_16x16x4_f32(
        /*neg_a=*/false, a, /*neg_b=*/false, b,
        /*c_mod=*/(short)0, cinit, /*reuse_a=*/false, /*reuse_b=*/false);

#pragma unroll
    for (int r = 0; r < 8; ++r) {
        bool pred = d[r] <= thresh;
        mask[r] = __builtin_amdgcn_ballot_w32(pred);
    }
}

// ---------------- Kernel 0: pad neighbor index region with -1 ----------------
__global__ void nbr_fill_kernel(int* __restrict__ p, int n) {
    int i = blockIdx.x * blockDim.x + threadIdx.x;
    if (i < n) p[i] = -1;
}

// ---------------- Kernel 1: per-query neighbor counts ----------------
__global__ void nbr_count_kernel(const float* __restrict__ data,
                                 const float* __restrict__ queries,
                                 const float* __restrict__ radius,
                                 int* __restrict__ counts,
                                 int n_data) {
    unsigned lane = threadIdx.x & 31u;
    int wave = threadIdx.x >> 5;
    int tile = blockIdx.x * WAVES_PER_BLOCK + wave;
    int row_base = tile * TILE;

    float r = radius[0];
    float r2 = r * r;

    v2f a; v8f cinit;
    load_query_tile(queries, row_base, lane, a, cinit);

    int cnt[16];
#pragma unroll
    for (int i = 0; i < 16; ++i) cnt[i] = 0;

    int ntiles = n_data / TILE;   // assumed even (16384/16 = 1024)

    // 2-deep software pipeline over data tiles.
    DTile t0 = dload(data, 0, lane);
    DTile t1 = dload(data, TILE, lane);
    for (int j = 0; j < ntiles; j += 2) {
        int j2 = (j + 2 < ntiles) ? (j + 2) : 0;   // wrap: harmless redundant load
        int j3 = (j + 3 < ntiles) ? (j + 3) : 1;
        DTile t2 = dload(data, j2 * TILE, lane);
        DTile t3 = dload(data, j3 * TILE, lane);

        unsigned mask[8];
        tile_masks(t0, lane, a, cinit, r2, mask);
#pragma unroll
        for (int r8 = 0; r8 < 8; ++r8) {
            cnt[r8]     += __popc(mask[r8] & 0xFFFFu);
            cnt[r8 + 8] += __popc(mask[r8] >> 16);
        }
        tile_masks(t1, lane, a, cinit, r2, mask);
#pragma unroll
        for (int r8 = 0; r8 < 8; ++r8) {
            cnt[r8]     += __popc(mask[r8] & 0xFFFFu);
            cnt[r8 + 8] += __popc(mask[r8] >> 16);
        }
        t0 = t2;
        t1 = t3;
    }

    if (lane < 16) {
        int v = 0;
#pragma unroll
        for (int i = 0; i < 16; ++i) v = (lane == (unsigned)i) ? cnt[i] : v;
        counts[row_base + (int)lane] = v;
    }
}

// ---------------- Kernel 2: exclusive scan -> splits[n+1] ----------------
__global__ void nbr_scan_kernel(const int* __restrict__ counts,
                                int* __restrict__ splits, int n) {
    __shared__ int sm[256];
    int t = threadIdx.x;
    int chunk = n / 256;
    int base = t * chunk;

    int sum = 0;
    for (int j = 0; j < chunk; ++j) sum += counts[base + j];
    sm[t] = sum;
    __syncthreads();

    for (int off = 1; off < 256; off <<= 1) {
        int v = (t >= off) ? sm[t - off] : 0;
        __syncthreads();
        sm[t] += v;
        __syncthreads();
    }

    int run = (t == 0) ? 0 : sm[t - 1];
    for (int j = 0; j < chunk; ++j) {
        splits[base + j] = run;
        run += counts[base + j];
    }
    if (t == 255) splits[n] = run;
}

// ---------------- Kernel 3: emit compacted column indices ----------------
__global__ void nbr_emit_kernel(const float* __restrict__ data,
                                const float* __restrict__ queries,
                                const float* __restrict__ radius,
                                const int* __restrict__ splits,
                                int* __restrict__ nbrs,
                                int n_data) {
    unsigned lane = threadIdx.x & 31u;
    int wave = threadIdx.x >> 5;
    int tile = blockIdx.x * WAVES_PER_BLOCK + wave;
    int row_base = tile * TILE;

    float r = radius[0];
    float r2 = r * r;

    v2f a; v8f cinit;
    load_query_tile(queries, row_base, lane, a, cinit);

    // Wave-uniform cursors: cur[i] = splits[row_base + i]
    int myv = 0;
    if (lane < 16) myv = splits[row_base + (int)lane];
    int cur[16];
#pragma unroll
    for (int i = 0; i < 16; ++i) cur[i] = __builtin_amdgcn_readlane(myv, i);

    unsigned lanebit = 1u << lane;
    unsigned halfmask = (lane < 16) ? 0xFFFFu : 0xFFFF0000u;
    unsigned prefix = (lanebit - 1u) & halfmask;
    int mycol = (int)(lane & 15);

    int ntiles = n_data / TILE;

    DTile t0 = dload(data, 0, lane);
    DTile t1 = dload(data, TILE, lane);
    for (int j = 0; j < ntiles; j += 2) {
        int j2 = (j + 2 < ntiles) ? (j + 2) : 0;
        int j3 = (j + 3 < ntiles) ? (j + 3) : 1;
        DTile t2 = dload(data, j2 * TILE, lane);
        DTile t3 = dload(data, j3 * TILE, lane);

#pragma unroll
        for (int half = 0; half < 2; ++half) {
            const DTile& tc = (half == 0) ? t0 : t1;
            int col_base = (j + half) * TILE;
            unsigned mask[8];
            tile_masks(tc, lane, a, cinit, r2, mask);
#pragma unroll
            for (int r8 = 0; r8 < 8; ++r8) {
                unsigned m = mask[r8];
                int curbase = (lane < 16) ? cur[r8] : cur[r8 + 8];
                if (m & lanebit) {
                    int rank = __popc(m & prefix);
                    int idx = curbase + rank;
                    if (idx < MAX_TOTAL) nbrs[idx] = col_base + mycol;
                }
                cur[r8]     += __popc(m & 0xFFFFu);
                cur[r8 + 8] += __popc(m >> 16);
            }
        }
        t0 = t2;
        t1 = t3;
    }
}

extern "C" void kernel_launch(void* const* d_in, const int* in_sizes, int n_in,
                              void* d_out, int out_size, void* d_ws, size_t ws_size,
                              hipStream_t stream) {
    const float* data    = (const float*)d_in[0];
    const float* queries = (const float*)d_in[1];
    const float* radius  = (const float*)d_in[2];

    int n_data    = in_sizes[0] / 3;   // 16384
    int n_queries = in_sizes[1] / 3;   // 16384

    int* outi   = (int*)d_out;
    int* nbrs   = outi;                // [MAX_TOTAL]
    int* splits = outi + MAX_TOTAL;    // [n_queries + 1]
    int* counts = (int*)d_ws;          // [n_queries]

    // 0) pad neighbor list with -1
    nbr_fill_kernel<<<(MAX_TOTAL + 255) / 256, 256, 0, stream>>>(nbrs, MAX_TOTAL);

    // 1) counts per query (WMMA distance tiles + ballot popcounts)
    int tiles = n_queries / TILE;                    // 1024
    int blocks = tiles / WAVES_PER_BLOCK;            // 128 blocks x 8 waves
    nbr_count_kernel<<<blocks, 32 * WAVES_PER_BLOCK, 0, stream>>>(
        data, queries, radius, counts, n_data);

    // 2) exclusive scan -> splits
    nbr_scan_kernel<<<1, 256, 0, stream>>>(counts, splits, n_queries);

    // 3) emit compacted indices
    nbr_emit_kernel<<<blocks, 32 * WAVES_PER_BLOCK, 0, stream>>>(
        data, queries, radius, splits, nbrs, n_data);
}